// CubeAttention_42202348650569
// MI455X (gfx1250) — compile-verified
//
#include <hip/hip_runtime.h>

#define GN 20
#define NV 8000        // 20^3 voxels

typedef float v2f __attribute__((ext_vector_type(2)));
typedef float v8f __attribute__((ext_vector_type(8)));

// ---------------------------------------------------------------------------
// Kernel 1: k_rel / v_rel  (125 x 64 each)
// p = o1*25 + o2*5 + o3 ; channels [0:32)=relpos_w[o1], [32:64)=relpos_w[o2],
// [64:96)=relpos_w[o3] matmul'ed against Wk/Wv rows [0:96).
// ---------------------------------------------------------------------------
__global__ void __launch_bounds__(64)
krel_vrel_kernel(const float* __restrict__ relpos_w,
                 const float* __restrict__ Wk, const float* __restrict__ bk,
                 const float* __restrict__ Wv, const float* __restrict__ bv,
                 float* __restrict__ krel, float* __restrict__ vrel) {
    int p = blockIdx.x;          // 0..124
    int c = threadIdx.x;         // 0..63
    int o1 = p / 25, o2 = (p / 5) % 5, o3 = p % 5;
    float k = bk[c], v = bv[c];
    #pragma unroll 4
    for (int j = 0; j < 32; ++j) {
        float r1 = relpos_w[o1 * 32 + j];
        float r2 = relpos_w[o2 * 32 + j];
        float r3 = relpos_w[o3 * 32 + j];
        k += r1 * Wk[(j     ) * 64 + c] + r2 * Wk[(32 + j) * 64 + c] + r3 * Wk[(64 + j) * 64 + c];
        v += r1 * Wv[(j     ) * 64 + c] + r2 * Wv[(32 + j) * 64 + c] + r3 * Wv[(64 + j) * 64 + c];
    }
    krel[p * 64 + c] = k;
    vrel[p * 64 + c] = v;
}

// ---------------------------------------------------------------------------
// Kernel 2: Q = SE @ Wq + bq ; QT = Q @ Wk_nb^T   (M = 16 voxels per wave)
// f32 WMMA 16x16x4. A layout: lane l holds row m=l&15, K-pair (4s+2*(l>>4)).
// B layout (mirror): lane l holds col n, same K-pair.
// C/D layout: lane l, vgpr r -> row m = r + 8*(l>>4), col n = 16t + (l&15).
// ---------------------------------------------------------------------------
__global__ void __launch_bounds__(128)
q_qt_kernel(const float* __restrict__ se, const float* __restrict__ Wq,
            const float* __restrict__ bq, const float* __restrict__ Wk,
            float* __restrict__ q_out, float* __restrict__ qt_out) {
    __shared__ float As[4][16 * 64];
    int tid  = threadIdx.x;
    int w    = tid >> 5;
    int lane = tid & 31;
    int lh   = lane >> 4;
    int l15  = lane & 15;
    int vbase = (blockIdx.x * 4 + w) * 16;

    // Stage A = se rows into this wave's LDS slice (coalesced)
    #pragma unroll
    for (int i = 0; i < 32; ++i)
        As[w][i * 32 + lane] = se[vbase * 64 + i * 32 + lane];
    __syncthreads();

    v8f acc[4];
    // ---- stage 1: Q = A @ Wq + bq ----
    #pragma unroll
    for (int t = 0; t < 4; ++t) {
        int n = t * 16 + l15;
        float bqv = bq[n];
        v8f c;
        #pragma unroll
        for (int r = 0; r < 8; ++r) c[r] = bqv;
        #pragma unroll
        for (int s = 0; s < 16; ++s) {
            int kb = s * 4 + lh * 2;
            v2f a, b;
            a.x = As[w][l15 * 64 + kb];
            a.y = As[w][l15 * 64 + kb + 1];
            b.x = Wq[kb * 64 + n];
            b.y = Wq[(kb + 1) * 64 + n];
            c = __builtin_amdgcn_wmma_f32_16x16x4_f32(false, a, false, b, (short)0, c, false, false);
        }
        acc[t] = c;
    }
    __syncthreads();   // everyone done reading As before overwrite
    // store Q to global and re-stage into LDS as next A operand
    #pragma unroll
    for (int t = 0; t < 4; ++t) {
        int n = t * 16 + l15;
        #pragma unroll
        for (int r = 0; r < 8; ++r) {
            int m = r + 8 * lh;
            q_out[(vbase + m) * 64 + n] = acc[t][r];
            As[w][m * 64 + n] = acc[t][r];
        }
    }
    __syncthreads();
    // ---- stage 2: QT = Q @ Wk_nb^T ; B[k][n] = Wk[(96+n)*64 + k] ----
    #pragma unroll
    for (int t = 0; t < 4; ++t) {
        int n = t * 16 + l15;
        v8f c;
        #pragma unroll
        for (int r = 0; r < 8; ++r) c[r] = 0.0f;
        #pragma unroll
        for (int s = 0; s < 16; ++s) {
            int kb = s * 4 + lh * 2;
            v2f a, b;
            a.x = As[w][l15 * 64 + kb];
            a.y = As[w][l15 * 64 + kb + 1];
            b.x = Wk[(96 + n) * 64 + kb];
            b.y = Wk[(96 + n) * 64 + kb + 1];
            c = __builtin_amdgcn_wmma_f32_16x16x4_f32(false, a, false, b, (short)0, c, false, false);
        }
        #pragma unroll
        for (int r = 0; r < 8; ++r)
            qt_out[(vbase + r + 8 * lh) * 64 + n] = c[r];
    }
}

// ---------------------------------------------------------------------------
// Kernel 3: per-voxel attention core.
// logits[p] = nb[p].qt + q.krel[p] - 1e9*(1-mask)  -> softmax ->
// snb = scores@nb ; svrel = scores@v_rel
// ---------------------------------------------------------------------------
__global__ void __launch_bounds__(128)
attn_kernel(const float* __restrict__ se, const float* __restrict__ krel,
            const float* __restrict__ vrel, const float* __restrict__ q,
            const float* __restrict__ qt,
            float* __restrict__ snb_out, float* __restrict__ svrel_out) {
    int v = blockIdx.x;
    int z = v % GN, y = (v / GN) % GN, x = v / (GN * GN);
    int tid = threadIdx.x;

    __shared__ float qs[64], qts[64];
    __shared__ float lg[128], red[128], sc[128];
    __shared__ int   nb_base[128];

    if (tid < 64) { qs[tid] = q[v * 64 + tid]; qts[tid] = qt[v * 64 + tid]; }
    __syncthreads();

    if (tid < 125) {
        int p = tid;
        int o1 = p / 25, o2 = (p / 5) % 5, o3 = p % 5;
        // data offsets: x uses o2, y uses o1, z uses o3 (reference's axis order)
        int nx = x + o2 - 2, ny = y + o1 - 2, nz = z + o3 - 2;
        bool ing = (nx >= 0) && (nx < GN) && (ny >= 0) && (ny < GN) && (nz >= 0) && (nz < GN);
        int base = ing ? ((nx * GN + ny) * GN + nz) * 64 : -1;
        nb_base[p] = base;
        float dn = 0.0f;
        if (ing) {
            const float4* nb4 = (const float4*)(se + base);
            const float4* qt4 = (const float4*)qts;
            #pragma unroll
            for (int d = 0; d < 16; ++d) {
                float4 a = nb4[d], b = qt4[d];
                dn += a.x * b.x + a.y * b.y + a.z * b.z + a.w * b.w;
            }
        }
        float dr = 0.0f;
        const float4* kr4 = (const float4*)(krel + p * 64);
        const float4* q4  = (const float4*)qs;
        #pragma unroll
        for (int d = 0; d < 16; ++d) {
            float4 a = kr4[d], b = q4[d];
            dr += a.x * b.x + a.y * b.y + a.z * b.z + a.w * b.w;
        }
        // mask: reference pairs o1 with x-validity, o2 with y-validity (swapped!)
        bool vx = (x + o1 > 2) && (x + o1 < 22);
        bool vy = (y + o2 > 2) && (y + o2 < 22);
        bool vz = (z + o3 > 2) && (z + o3 < 22);
        lg[p] = dn + dr - ((vx && vy && vz) ? 0.0f : 1.0e9f);
    } else {
        lg[tid] = -3.0e38f;
        nb_base[tid] = -1;
    }
    __syncthreads();

    // max-reduce
    red[tid] = lg[tid];
    __syncthreads();
    for (int off = 64; off > 0; off >>= 1) {
        if (tid < off) red[tid] = fmaxf(red[tid], red[tid + off]);
        __syncthreads();
    }
    float mx = red[0];
    __syncthreads();

    float e = (tid < 125) ? __expf(lg[tid] - mx) : 0.0f;
    sc[tid]  = e;
    red[tid] = e;
    __syncthreads();
    for (int off = 64; off > 0; off >>= 1) {
        if (tid < off) red[tid] += red[tid + off];
        __syncthreads();
    }
    float inv = 1.0f / red[0];
    __syncthreads();
    sc[tid] *= inv;
    __syncthreads();

    if (tid < 64) {
        int d = tid;
        float acc = 0.0f;
        for (int p = 0; p < 125; ++p) {
            int b = nb_base[p];
            if (b >= 0) acc += sc[p] * se[b + d];
        }
        snb_out[v * 64 + d] = acc;
    } else {
        int c = tid - 64;
        float acc = 0.0f;
        for (int p = 0; p < 125; ++p)
            acc += sc[p] * vrel[p * 64 + c];
        svrel_out[v * 64 + c] = acc;
    }
}

// ---------------------------------------------------------------------------
// Kernel 4: OUT = (SNB @ Wv_nb + SVREL) @ Wo + bo   (WMMA, same wave layout)
// ---------------------------------------------------------------------------
__global__ void __launch_bounds__(128)
out_kernel(const float* __restrict__ snb, const float* __restrict__ svrel,
           const float* __restrict__ Wv, const float* __restrict__ Wo,
           const float* __restrict__ bo, float* __restrict__ out) {
    __shared__ float As[4][16 * 64];
    int tid  = threadIdx.x;
    int w    = tid >> 5;
    int lane = tid & 31;
    int lh   = lane >> 4;
    int l15  = lane & 15;
    int vbase = (blockIdx.x * 4 + w) * 16;

    #pragma unroll
    for (int i = 0; i < 32; ++i)
        As[w][i * 32 + lane] = snb[vbase * 64 + i * 32 + lane];
    __syncthreads();

    v8f acc[4];
    // ---- stage 1: T = SNB @ Wv_nb + SVREL ; B[k][n] = Wv[(96+k)*64 + n] ----
    #pragma unroll
    for (int t = 0; t < 4; ++t) {
        int n = t * 16 + l15;
        v8f c;
        #pragma unroll
        for (int r = 0; r < 8; ++r)
            c[r] = svrel[(vbase + r + 8 * lh) * 64 + n];
        #pragma unroll
        for (int s = 0; s < 16; ++s) {
            int kb = s * 4 + lh * 2;
            v2f a, b;
            a.x = As[w][l15 * 64 + kb];
            a.y = As[w][l15 * 64 + kb + 1];
            b.x = Wv[(96 + kb) * 64 + n];
            b.y = Wv[(96 + kb + 1) * 64 + n];
            c = __builtin_amdgcn_wmma_f32_16x16x4_f32(false, a, false, b, (short)0, c, false, false);
        }
        acc[t] = c;
    }
    __syncthreads();
    #pragma unroll
    for (int t = 0; t < 4; ++t) {
        int n = t * 16 + l15;
        #pragma unroll
        for (int r = 0; r < 8; ++r)
            As[w][(r + 8 * lh) * 64 + n] = acc[t][r];
    }
    __syncthreads();
    // ---- stage 2: OUT = T @ Wo + bo ----
    #pragma unroll
    for (int t = 0; t < 4; ++t) {
        int n = t * 16 + l15;
        float bov = bo[n];
        v8f c;
        #pragma unroll
        for (int r = 0; r < 8; ++r) c[r] = bov;
        #pragma unroll
        for (int s = 0; s < 16; ++s) {
            int kb = s * 4 + lh * 2;
            v2f a, b;
            a.x = As[w][l15 * 64 + kb];
            a.y = As[w][l15 * 64 + kb + 1];
            b.x = Wo[kb * 64 + n];
            b.y = Wo[(kb + 1) * 64 + n];
            c = __builtin_amdgcn_wmma_f32_16x16x4_f32(false, a, false, b, (short)0, c, false, false);
        }
        #pragma unroll
        for (int r = 0; r < 8; ++r)
            out[(vbase + r + 8 * lh) * 64 + n] = c[r];
    }
}

// ---------------------------------------------------------------------------
extern "C" void kernel_launch(void* const* d_in, const int* in_sizes, int n_in,
                              void* d_out, int out_size, void* d_ws, size_t ws_size,
                              hipStream_t stream) {
    const float* se  = (const float*)d_in[0];
    const float* rpw = (const float*)d_in[1];
    const float* Wq  = (const float*)d_in[2];
    const float* bq  = (const float*)d_in[3];
    const float* Wk  = (const float*)d_in[4];
    const float* bk  = (const float*)d_in[5];
    const float* Wv  = (const float*)d_in[6];
    const float* bv  = (const float*)d_in[7];
    const float* Wo  = (const float*)d_in[8];
    const float* bo  = (const float*)d_in[9];

    float* ws    = (float*)d_ws;
    float* krel  = ws;                  // 125*64
    float* vrel  = ws + 8000;           // 125*64
    float* q     = ws + 16000;          // 8000*64
    float* qt    = q + NV * 64;         // 8000*64
    float* snb   = qt + NV * 64;        // 8000*64
    float* svrel = snb + NV * 64;       // 8000*64
    float* out   = (float*)d_out;

    hipLaunchKernelGGL(krel_vrel_kernel, dim3(125), dim3(64), 0, stream,
                       rpw, Wk, bk, Wv, bv, krel, vrel);
    hipLaunchKernelGGL(q_qt_kernel, dim3(125), dim3(128), 0, stream,
                       se, Wq, bq, Wk, q, qt);
    hipLaunchKernelGGL(attn_kernel, dim3(NV), dim3(128), 0, stream,
                       se, krel, vrel, q, qt, snb, svrel);
    hipLaunchKernelGGL(out_kernel, dim3(125), dim3(128), 0, stream,
                       snb, svrel, Wv, Wo, bo, out);
}